// WeightNet_DW_39127152066739
// MI455X (gfx1250) — compile-verified
//
#include <hip/hip_runtime.h>
#include <math.h>

// Problem constants (match reference)
#define Bn   32
#define Cn   256
#define Hn   56
#define Wn   56
#define GAPn 16
#define Kn   3
#define PLANE (Hn * Wn)          // 3136
#define PLANE4 (PLANE / 4)       // 784 float4 per plane
#define ROW4  (Wn / 4)           // 14 float4 per row

typedef __attribute__((ext_vector_type(2))) float v2f;
typedef __attribute__((ext_vector_type(8))) float v8f;

// ---------------------------------------------------------------------------
// Kernel 1: global average pool.  One block per (b,c) plane.
// ---------------------------------------------------------------------------
__global__ void gap_kernel(const float* __restrict__ x, float* __restrict__ gap) {
    const int bc  = blockIdx.x;                 // 0 .. B*C-1
    const int tid = threadIdx.x;                // 0 .. 255
    const float4* xp = (const float4*)(x + (size_t)bc * PLANE);

    float sum = 0.0f;
    for (int v = tid; v < PLANE4; v += 256) {
        float4 d = xp[v];
        sum += (d.x + d.y) + (d.z + d.w);
    }

    __shared__ float red[256];
    red[tid] = sum;
    __syncthreads();
    for (int s = 128; s > 0; s >>= 1) {
        if (tid < s) red[tid] += red[tid + s];
        __syncthreads();
    }
    if (tid == 0) gap[bc] = red[0] * (1.0f / (float)PLANE);
}

// ---------------------------------------------------------------------------
// Kernel 2: gating network with fp32 WMMA (V_WMMA_F32_16X16X4_F32).
//   g  = gap @ reduce_w^T + reduce_b      [32,16]   (K = 256)
//   a  = sigmoid(g @ fc1_w^T)             [32,256]  (K = 16)
//   wk = a * fc2_w + fc2_b                [32,256,9]
// Single workgroup of 8 wave32s.  All WMMA-executing branches are
// wave-uniform, so EXEC is all-ones inside every WMMA.
// ---------------------------------------------------------------------------
__global__ void gate_wmma_kernel(const float* __restrict__ gap,
                                 const float* __restrict__ reduce_w,   // [16,256]
                                 const float* __restrict__ reduce_b,   // [16]
                                 const float* __restrict__ fc1_w,      // [256,16]
                                 const float* __restrict__ fc2_w,      // [256,9]
                                 const float* __restrict__ fc2_b,      // [256,9]
                                 float* __restrict__ wk)               // [32,256,9]
{
    __shared__ float s_g[Bn * GAPn];            // g matrix [32][16]

    const int tid   = threadIdx.x;
    const int wave  = tid >> 5;
    const int lane  = tid & 31;
    const int l15   = lane & 15;                // M (A frag) or N (B/C frag)
    const int khalf = (lane >> 4) * 2;          // lanes 0-15 -> K 0,1 ; 16-31 -> K 2,3
    const int rbase = (lane < 16) ? 0 : 8;      // C/D: VGPR r -> row r or r+8

    // ---- stage 1: g = gap @ reduce_w^T + reduce_b   (two 16-row M tiles) ----
    if (wave < 2) {
        const int mt = wave;                    // M tile 0 / 1
        v8f acc = {};
        for (int k0 = 0; k0 < Cn; k0 += 4) {
            v2f a, b;
            // A (16x4): A[m,k] = gap[mt*16+m, k]
            a.x = gap[(mt * 16 + l15) * Cn + k0 + khalf];
            a.y = gap[(mt * 16 + l15) * Cn + k0 + khalf + 1];
            // B (4x16): B[k,n] = reduce_w[n, k]
            b.x = reduce_w[l15 * Cn + k0 + khalf];
            b.y = reduce_w[l15 * Cn + k0 + khalf + 1];
            acc = __builtin_amdgcn_wmma_f32_16x16x4_f32(
                false, a, false, b, (short)0, acc, false, false);
        }
        const float bias = reduce_b[l15];
        #pragma unroll
        for (int r = 0; r < 8; ++r)
            s_g[(mt * 16 + rbase + r) * GAPn + l15] = acc[r] + bias;
    }
    __syncthreads();

    // ---- stage 2: a = sigmoid(g @ fc1_w^T); wk = a*fc2_w + fc2_b ----
    // 32 output tiles (2 M-tiles x 16 N-tiles); 4 tiles per wave.
    #pragma unroll
    for (int i = 0; i < 4; ++i) {
        const int tile = wave * 4 + i;
        const int mt = tile >> 4;               // 0..1
        const int nt = tile & 15;               // 0..15
        v8f acc = {};
        #pragma unroll
        for (int k0 = 0; k0 < GAPn; k0 += 4) {
            v2f a, b;
            a.x = s_g[(mt * 16 + l15) * GAPn + k0 + khalf];
            a.y = s_g[(mt * 16 + l15) * GAPn + k0 + khalf + 1];
            // B[k,n] = fc1_w[nt*16+n, k]
            b.x = fc1_w[(nt * 16 + l15) * GAPn + k0 + khalf];
            b.y = fc1_w[(nt * 16 + l15) * GAPn + k0 + khalf + 1];
            acc = __builtin_amdgcn_wmma_f32_16x16x4_f32(
                false, a, false, b, (short)0, acc, false, false);
        }
        const int c = nt * 16 + l15;            // channel
        #pragma unroll
        for (int r = 0; r < 8; ++r) {
            const int bb = mt * 16 + rbase + r; // batch
            const float av = 1.0f / (1.0f + __expf(-acc[r]));
            float* wkp = wk + ((size_t)bb * Cn + c) * (Kn * Kn);
            const float* fwp = fc2_w + c * (Kn * Kn);
            const float* fbp = fc2_b + c * (Kn * Kn);
            #pragma unroll
            for (int t = 0; t < Kn * Kn; ++t)
                wkp[t] = fmaf(av, fwp[t], fbp[t]);
        }
    }
}

// ---------------------------------------------------------------------------
// Kernel 3: dynamic depthwise 3x3 conv.  One block per (b,c) plane.
// Plane staged in LDS with zero halo; padded row stride 64 floats keeps
// float4 alignment and spreads banks.
// ---------------------------------------------------------------------------
#define LW 64                                   // LDS row stride (floats)
#define LH 58                                   // 56 + halo

__global__ void dwconv_kernel(const float* __restrict__ x,
                              const float* __restrict__ wk,
                              float* __restrict__ out)
{
    __shared__ float tile[LH * LW];             // 14848 B

    const int bc  = blockIdx.x;
    const int tid = threadIdx.x;
    const float4* xp = (const float4*)(x + (size_t)bc * PLANE);
    float4*       op = (float4*)(out + (size_t)bc * PLANE);

    // per-plane 3x3 taps (same 9 floats for every thread in the block)
    const float* kp = wk + (size_t)bc * (Kn * Kn);
    float kt[9];
    #pragma unroll
    for (int t = 0; t < 9; ++t) kt[t] = kp[t];

    // zero LDS (gives us the zero halo for free)
    for (int i = tid; i < LH * LW; i += 256) tile[i] = 0.0f;
    __syncthreads();

    // stage interior: float4 v covers row h = v/14, cols 4*(v%14) .. +3.
    // interior pixel (h,w) lives at tile[(h+1)*LW + w + 4]  (16B aligned).
    for (int v = tid; v < PLANE4; v += 256) {
        float4 d = xp[v];
        const int h = v / ROW4;
        const int w = (v - h * ROW4) * 4;
        *(float4*)&tile[(h + 1) * LW + w + 4] = d;
    }
    __syncthreads();

    // compute 4 outputs per float4 slot, branch-free via the halo
    for (int v = tid; v < PLANE4; v += 256) {
        const int h = v / ROW4;
        const int w = (v - h * ROW4) * 4;
        float r0 = 0.f, r1 = 0.f, r2 = 0.f, r3 = 0.f;
        #pragma unroll
        for (int dy = 0; dy < 3; ++dy) {
            const float* row = &tile[(h + dy) * LW + w + 3];  // input cols w-1 .. w+4
            const float c0 = row[0], c1 = row[1], c2 = row[2];
            const float c3 = row[3], c4 = row[4], c5 = row[5];
            const float ka = kt[dy * 3 + 0];
            const float kb = kt[dy * 3 + 1];
            const float kc = kt[dy * 3 + 2];
            r0 = fmaf(c0, ka, fmaf(c1, kb, fmaf(c2, kc, r0)));
            r1 = fmaf(c1, ka, fmaf(c2, kb, fmaf(c3, kc, r1)));
            r2 = fmaf(c2, ka, fmaf(c3, kb, fmaf(c4, kc, r2)));
            r3 = fmaf(c3, ka, fmaf(c4, kb, fmaf(c5, kc, r3)));
        }
        op[v] = make_float4(r0, r1, r2, r3);
    }
}

// ---------------------------------------------------------------------------
// Launch
// ---------------------------------------------------------------------------
extern "C" void kernel_launch(void* const* d_in, const int* in_sizes, int n_in,
                              void* d_out, int out_size, void* d_ws, size_t ws_size,
                              hipStream_t stream) {
    const float* x        = (const float*)d_in[0];   // [32,256,56,56]
    const float* reduce_w = (const float*)d_in[1];   // [16,256]
    const float* reduce_b = (const float*)d_in[2];   // [16]
    const float* fc1_w    = (const float*)d_in[3];   // [256,16]
    const float* fc2_w    = (const float*)d_in[4];   // [256,9]
    const float* fc2_b    = (const float*)d_in[5];   // [256,9]
    float* out = (float*)d_out;

    // workspace layout: gap [32*256] | wk [32*256*9]
    float* gap = (float*)d_ws;
    float* wk  = gap + Bn * Cn;

    gap_kernel<<<Bn * Cn, 256, 0, stream>>>(x, gap);
    gate_wmma_kernel<<<1, 256, 0, stream>>>(gap, reduce_w, reduce_b,
                                            fc1_w, fc2_w, fc2_b, wk);
    dwconv_kernel<<<Bn * Cn, 256, 0, stream>>>(x, wk, out);
}